// CmapEncdoer_5153960755827
// MI455X (gfx1250) — compile-verified
//
#include <hip/hip_runtime.h>
#include <hip/hip_bf16.h>

// GCN encoder (mu, logstd) fused: one GEMM x@[W1|W2] (f32 WMMA 16x16x4),
// one 400-channel degree-normalized aggregation pass with f32 atomics.

typedef float v2f __attribute__((ext_vector_type(2)));
typedef float v8f __attribute__((ext_vector_type(8)));

#define IN_C   128
#define OUT_C  200
#define CC     (2*OUT_C)    // 400 fused channels
#define NT     (CC/16)      // 25 col tiles
#define LDSPAD 132          // 132 mod 64 = 4 -> conflict-free ds_load_b64 across 16 rows

// ---------------- degree / norm ----------------
__global__ void deg_init_kernel(int* __restrict__ deg, long long n) {
    long long i = blockIdx.x * (long long)blockDim.x + threadIdx.x;
    if (i < n) deg[i] = 1;  // self loop
}

__global__ void deg_edge_kernel(const int* __restrict__ ei, int* __restrict__ deg,
                                long long E) {
    long long e = blockIdx.x * (long long)blockDim.x + threadIdx.x;
    if (e < E) atomicAdd(&deg[ei[E + e]], 1);  // col = dst
}

__global__ void dinv_kernel(const int* __restrict__ deg, float* __restrict__ dinv,
                            long long n) {
    long long i = blockIdx.x * (long long)blockDim.x + threadIdx.x;
    if (i < n) dinv[i] = rsqrtf((float)deg[i]);  // deg >= 1 always
}

// -------- pack W = [W1 | W2] into K-pair interleaved layout ----------
// Wpk[(k>>1)*CC*2 + n*2 + (k&1)]  ->  the (k,k+1) pair for col n is one b64.
__global__ void packw_kernel(const float* __restrict__ W1, const float* __restrict__ W2,
                             float* __restrict__ Wpk) {
    int i = blockIdx.x * blockDim.x + threadIdx.x;  // 128*400 = 51200
    if (i >= IN_C * CC) return;
    int k = i / CC, n = i - k * CC;
    float v = (n < OUT_C) ? W1[k * OUT_C + n] : W2[k * OUT_C + (n - OUT_C)];
    Wpk[((size_t)(k >> 1) * CC + n) * 2 + (k & 1)] = v;
}

// ---------------- WMMA GEMM: xws[r][n] = dinv[r] * sum_k x[r][k]*Wpk[k][n] -------
// Block = 128 threads (4 waves). Each block: one 16-row tile, waves split 25 col tiles.
__global__ __launch_bounds__(128) void gemm_xw_kernel(const float* __restrict__ x,
                                                      const float* __restrict__ Wpk,
                                                      const float* __restrict__ dinv,
                                                      float* __restrict__ xws,
                                                      long long N) {
    __shared__ float lx[16 * LDSPAD];
    const int t = threadIdx.x;
    const long long row0 = (long long)blockIdx.x * 16;

    // Stage 16x128 x-tile into LDS: b128 global loads, b64 LDS stores (8448B).
    #pragma unroll
    for (int j = 0; j < 4; ++j) {
        int flat = (j * 128 + t) * 4;          // float index within tile
        int r = flat >> 7;                     // /128
        int c = flat & 127;
        long long gr = row0 + r; if (gr >= N) gr = N - 1;   // tail clamp
        float4 v = *(const float4*)(x + gr * IN_C + c);
        float* dst = &lx[r * LDSPAD + c];
        ((float2*)dst)[0] = make_float2(v.x, v.y);
        ((float2*)dst)[1] = make_float2(v.z, v.w);
    }
    __syncthreads();

    const int wave = t >> 5;
    const int lane = t & 31;
    const int half = lane >> 4;   // K sub-pair select (WMMA f32 A/B layout)
    const int l16  = lane & 15;
    const bool fullTile = (row0 + 16 <= N);

    // dinv for the 8 accumulator rows this lane covers (same for every col tile)
    float dv[8];
    #pragma unroll
    for (int j = 0; j < 8; ++j) {
        long long gr = row0 + j + (half << 3); if (gr >= N) gr = N - 1;
        dv[j] = dinv[gr];
    }

    for (int nt = wave; nt < NT; nt += 4) {
        const int n = nt * 16 + l16;
        // B base for this lane/col: pairs (2*kk+half) stride CC*2 floats apart.
        const float* bptr = Wpk + ((size_t)half * CC + n) * 2;
        v8f acc = {0.f, 0.f, 0.f, 0.f, 0.f, 0.f, 0.f, 0.f};
        #pragma unroll 8
        for (int kk = 0; kk < IN_C / 4; ++kk) {
            const int k = 4 * kk + 2 * half;
            // A 16x4: lanes0-15 rows M=l16 hold K={4kk,4kk+1}; lanes16-31 K={4kk+2,4kk+3}
            const float2 av = *(const float2*)&lx[l16 * LDSPAD + k];
            v2f a; a[0] = av.x; a[1] = av.y;
            // B 4x16: one b64 per K-step, coalesced across lanes, imm offsets.
            const float2 bv = *(const float2*)(bptr + (size_t)kk * (2 * CC * 2));
            v2f b; b[0] = bv.x; b[1] = bv.y;
            acc = __builtin_amdgcn_wmma_f32_16x16x4_f32(
                false, a, false, b, (short)0, acc, false, false);
        }
        // C/D layout: VGPR j -> row M=j (lanes0-15) / M=j+8 (lanes16-31), col N=l16
        float* op = xws + (row0 + (half << 3)) * CC + n;
        if (fullTile) {
            #pragma unroll
            for (int j = 0; j < 8; ++j) op[(size_t)j * CC] = acc[j] * dv[j];
        } else {
            #pragma unroll
            for (int j = 0; j < 8; ++j) {
                if (row0 + j + (half << 3) < N) op[(size_t)j * CC] = acc[j] * dv[j];
            }
        }
    }
}

// ---------------- self-loop + bias: out[v] = dinv[v]*xws[v] + b ----------------
__global__ void selfloop_kernel(const float* __restrict__ xws, const float* __restrict__ dinv,
                                const float* __restrict__ b1, const float* __restrict__ b2,
                                float* __restrict__ out, long long N) {
    long long tid = blockIdx.x * (long long)blockDim.x + threadIdx.x;
    long long total = N * (CC / 4);
    if (tid >= total) return;
    long long v = tid / (CC / 4);
    int c4 = (int)(tid - v * (CC / 4)) * 4;
    float dvv = dinv[v];
    float4 m = *(const float4*)(xws + v * CC + c4);
    float4 bb; float* o;
    if (c4 < OUT_C) { bb = *(const float4*)(b1 + c4);          o = out + v * OUT_C + c4; }
    else            { bb = *(const float4*)(b2 + (c4 - OUT_C)); o = out + (N + v) * OUT_C + (c4 - OUT_C); }
    o[0] = m.x * dvv + bb.x;
    o[1] = m.y * dvv + bb.y;
    o[2] = m.z * dvv + bb.z;
    o[3] = m.w * dvv + bb.w;
}

// ---------------- edge scatter: out[v] += dinv[v] * xws[u] ----------------
__global__ void edge_scatter_kernel(const int* __restrict__ ei, const float* __restrict__ xws,
                                    const float* __restrict__ dinv, float* __restrict__ out,
                                    long long E, long long N) {
    long long tid = blockIdx.x * (long long)blockDim.x + threadIdx.x;
    long long total = E * (CC / 4);
    if (tid >= total) return;
    long long e = tid / (CC / 4);
    int c4 = (int)(tid - e * (CC / 4)) * 4;
    int u = ei[e];           // row = src
    int v = ei[E + e];       // col = dst
    float dvv = dinv[v];
    float4 m = *(const float4*)(xws + (size_t)u * CC + c4);
    float* o = (c4 < OUT_C) ? (out + (size_t)v * OUT_C + c4)
                            : (out + ((size_t)N + v) * OUT_C + (c4 - OUT_C));
    atomicAdd(o + 0, m.x * dvv);
    atomicAdd(o + 1, m.y * dvv);
    atomicAdd(o + 2, m.z * dvv);
    atomicAdd(o + 3, m.w * dvv);
}

extern "C" void kernel_launch(void* const* d_in, const int* in_sizes, int n_in,
                              void* d_out, int out_size, void* d_ws, size_t ws_size,
                              hipStream_t stream) {
    const float* x  = (const float*)d_in[0];
    const int*   ei = (const int*)d_in[1];   // edge_index [2][E], int per bridge doc
    const float* W1 = (const float*)d_in[2];
    const float* b1 = (const float*)d_in[3];
    const float* W2 = (const float*)d_in[4];
    const float* b2 = (const float*)d_in[5];
    float* out = (float*)d_out;

    const long long N = in_sizes[0] / IN_C;
    const long long E = in_sizes[1] / 2;

    // workspace layout
    char* w = (char*)d_ws;
    float* xws  = (float*)w;                                  // N*400 f32
    size_t off  = (size_t)N * CC * sizeof(float);
    float* dinv = (float*)(w + off);                          // N f32
    int*   deg  = (int*)(w + off + (size_t)N * 4);            // N i32
    float* Wpk  = (float*)(w + off + (size_t)N * 8);          // 128*400 f32 (pair-packed)

    const int B = 256;
    deg_init_kernel<<<(unsigned)((N + B - 1) / B), B, 0, stream>>>(deg, N);
    deg_edge_kernel<<<(unsigned)((E + B - 1) / B), B, 0, stream>>>(ei, deg, E);
    dinv_kernel<<<(unsigned)((N + B - 1) / B), B, 0, stream>>>(deg, dinv, N);
    packw_kernel<<<(IN_C * CC + B - 1) / B, B, 0, stream>>>(W1, W2, Wpk);

    gemm_xw_kernel<<<(unsigned)((N + 15) / 16), 128, 0, stream>>>(x, Wpk, dinv, xws, N);

    long long selfTotal = N * (CC / 4);
    selfloop_kernel<<<(unsigned)((selfTotal + B - 1) / B), B, 0, stream>>>(xws, dinv, b1, b2, out, N);

    long long edgeTotal = E * (CC / 4);
    edge_scatter_kernel<<<(unsigned)((edgeTotal + B - 1) / B), B, 0, stream>>>(ei, xws, dinv, out, E, N);
}